// TopLeftPool_326417514953
// MI455X (gfx1250) — compile-verified
//
#include <hip/hip_runtime.h>
#include <stdint.h>

// Shapes from the reference: x (16,128,128,128) f32, guide (16,1,128,128) f32.
#define NB    16
#define NC    128
#define NH    128
#define NW    128
#define PLANE (NH * NW)   // 16384 elements = 64 KB per (b,c) plane

typedef float    __attribute__((ext_vector_type(4))) f32x4;
typedef uint32_t __attribute__((ext_vector_type(4))) u32x4;

// ---------------------------------------------------------------------------
// Kernel A1: guided cumulative max down each column (axis H).
// One block per batch b, thread j owns column j. Loads are independent of the
// running-max chain -> compiler clause-batches them (verified in round 1).
// Tie rule: strictly-greater replaces, so the earliest i wins (matches the
// reference combiner take_b = gb > ga).
// Outputs (to scratch): g1[b,i,j] = cummax, s1[b,i,j] = argmax_row*128 + j.
// ---------------------------------------------------------------------------
__global__ __launch_bounds__(NW)
void colscan_kernel(const float* __restrict__ guide,
                    float* __restrict__ g1,
                    uint32_t* __restrict__ s1) {
    const int b = blockIdx.x;
    const int j = threadIdx.x;
    const float* gb = guide + (size_t)b * PLANE;
    float*     g1b = g1 + (size_t)b * PLANE;
    uint32_t*  s1b = s1 + (size_t)b * PLANE;

    float best = -__builtin_huge_valf();
    int bestr = 0;
#pragma unroll 8
    for (int i = 0; i < NH; ++i) {
        float gv = gb[i * NW + j];               // coalesced
        if (gv > best) { best = gv; bestr = i; } // strict > : earliest i wins
        g1b[i * NW + j] = best;                  // coalesced
        s1b[i * NW + j] = (uint32_t)(bestr * NW + j);
    }
}

// ---------------------------------------------------------------------------
// Kernel A2: guided cumulative max along each row (axis W) of g1.
// One thread per (b,i) row; 2048 threads total. g1/s1 are L2-resident (2MB).
// Output: packed[b,i,j] = r[b, i, j*] * 128 + j*  (final gather index).
// ---------------------------------------------------------------------------
__global__ __launch_bounds__(256)
void rowscan_kernel(const float* __restrict__ g1,
                    const uint32_t* __restrict__ s1,
                    uint32_t* __restrict__ packed) {
    const int rid = blockIdx.x * 256 + threadIdx.x;  // 0 .. NB*NH-1
    const int b = rid >> 7;
    const int i = rid & (NH - 1);
    const float*    row_g = g1 + (size_t)b * PLANE + (size_t)i * NW;
    const uint32_t* row_s = s1 + (size_t)b * PLANE + (size_t)i * NW;
    uint32_t*       row_p = packed + (size_t)b * PLANE + (size_t)i * NW;

    float best = -__builtin_huge_valf();
    uint32_t bests = 0;
#pragma unroll 8
    for (int j = 0; j < NW; ++j) {
        float gv = row_g[j];
        uint32_t sv = row_s[j];
        if (gv > best) { best = gv; bests = sv; } // strict > : earliest j wins
        row_p[j] = bests;
    }
}

// ---------------------------------------------------------------------------
// Kernel B: the bandwidth-dominant gather (268 MB of HBM traffic).
// One block per (b,c) plane:
//   1) stage the 64 KB x-plane into LDS via GLOBAL_LOAD_ASYNC_TO_LDS_B128
//      (ASYNCcnt path, no VGPR staging) with TH_LOAD_NT — each plane is read
//      exactly once, don't let it evict the L2-resident index maps;
//   2) s_wait_asynccnt 0 per wave + workgroup barrier;
//   3) gather 4-wide: b128 packed-index loads (L2 hits, reused by all 128
//      channels), 4x ds_load_b32 LDS gathers, nontemporal b128 output stores
//      (output is write-once; keep L2 for x tiles + indices).
// ---------------------------------------------------------------------------
__global__ __launch_bounds__(256)
void gather_kernel(const float* __restrict__ x,
                   const uint32_t* __restrict__ packed,
                   float* __restrict__ out) {
    __shared__ __align__(16) float tile[PLANE];  // 64 KB

    const int p = blockIdx.x;        // plane id = b*NC + c
    const int b = p >> 7;            // NC == 128
    const int t = threadIdx.x;

    const float* xp = x + (size_t)p * PLANE;

    // LDS byte address: flat pointers to LDS carry the wave-relative LDS
    // offset in their low 32 bits (aperture layout), which is what
    // GLOBAL_LOAD_ASYNC_TO_LDS_B128 expects in its VDST register.
    uint32_t lds_base = (uint32_t)(uintptr_t)(&tile[0]);
    uint64_t gbase    = (uint64_t)(uintptr_t)xp;

#pragma unroll
    for (int k = 0; k < 16; ++k) {
        uint32_t off      = (uint32_t)(k * 256 + t) * 16u; // 16B granularity
        uint32_t lds_addr = lds_base + off;
        uint64_t gaddr    = gbase + off;
        asm volatile("global_load_async_to_lds_b128 %0, %1, off th:TH_LOAD_NT"
                     :: "v"(lds_addr), "v"(gaddr)
                     : "memory");
    }
    // Each wave drains its own ASYNCcnt; the workgroup barrier then makes the
    // whole tile visible to every wave.
    asm volatile("s_wait_asynccnt 0x0" ::: "memory");
    __syncthreads();

    const u32x4* pk4 = (const u32x4*)(packed + (size_t)b * PLANE);
    f32x4*       op4 = (f32x4*)(out + (size_t)p * PLANE);

    // PLANE/4 = 4096 vec4 elements; 256 threads -> 16 iterations.
#pragma unroll 4
    for (int k = 0; k < 16; ++k) {
        int m4 = k * 256 + t;
        u32x4 s = pk4[m4];               // b128 load, coalesced, L2 hit
        f32x4 v;
        v.x = tile[s.x];                 // 4x ds_load_b32 gather
        v.y = tile[s.y];
        v.z = tile[s.z];
        v.w = tile[s.w];
        __builtin_nontemporal_store(v, &op4[m4]);  // b128 store, TH_STORE_NT
    }
}

// ---------------------------------------------------------------------------
// Host-side launcher. d_in[0] = x, d_in[1] = guide (setup_inputs order).
// Scratch layout in d_ws: [g1 f32 1MB][s1 u32 1MB][packed u32 1MB] = 3 MB.
// ---------------------------------------------------------------------------
extern "C" void kernel_launch(void* const* d_in, const int* in_sizes, int n_in,
                              void* d_out, int out_size, void* d_ws, size_t ws_size,
                              hipStream_t stream) {
    (void)in_sizes; (void)n_in; (void)out_size; (void)ws_size;

    const float* x     = (const float*)d_in[0];
    const float* guide = (const float*)d_in[1];
    float* out = (float*)d_out;

    const size_t idx_elems = (size_t)NB * PLANE;           // 262144
    float*    g1     = (float*)d_ws;
    uint32_t* s1     = (uint32_t*)((char*)d_ws + idx_elems * sizeof(float));
    uint32_t* packed = (uint32_t*)((char*)d_ws + 2 * idx_elems * sizeof(float));

    // Stage 1: column scans (tiny).
    colscan_kernel<<<NB, NW, 0, stream>>>(guide, g1, s1);
    // Stage 2: row scans (tiny).
    rowscan_kernel<<<(NB * NH) / 256, 256, 0, stream>>>(g1, s1, packed);
    // Stage 3: bandwidth-bound gather via async global->LDS staging.
    gather_kernel<<<NB * NC, 256, 0, stream>>>(x, packed, out);
}